// ParallelModel_36206574305602
// MI455X (gfx1250) — compile-verified
//
#include <hip/hip_runtime.h>
#include <math.h>

// ---------------------------------------------------------------------------
// GNN forward (GIN x3 + TransformerConv x2 + pooled readout) for MI455X/gfx1250.
// Dense GEMMs use V_WMMA_F32_16X16X4_F32 (full fp32 tensor op, wave32).
// Each wave computes a 16x64 output strip (4 WMMA accumulators); one A
// fragment feeds 4 WMMA ops per k-step. B loads use 4 independent registers
// so the loads clause together and drain while WMMAs execute; all loop
// addressing is stride-bumped pointers (no per-iteration 64-bit multiply).
//
// Assumed d_in order = setup_inputs() insertion order, recursively, w before b:
//  0 x[25000*128] 1 edge_attr[400000*16] 2 edge_index[2*400000](int)
//  3 batch[25000](int)
//  4..15  gin{0,1,2}.{lin1.w,lin1.b,lin2.w,lin2.b}
//  16..25 gt0.{q.w,q.b,k.w,k.b,v.w,v.b,e.w,skip.w,skip.b,beta.w}
//  26..35 gt1 same
//  36 read1.w[192*64] 37 read1.b[64] 38 read2.w[64] 39 read2.b[1]
// ---------------------------------------------------------------------------

typedef float v2f __attribute__((ext_vector_type(2)));
typedef float v8f __attribute__((ext_vector_type(8)));

#define NFEAT 128
#define HC    128   // HEADS*HIDDEN
#define HID   64
#define NGRAPH 64

__device__ __forceinline__ v8f wmma_f32(v2f a, v2f b, v8f c) {
    return __builtin_amdgcn_wmma_f32_16x16x4_f32(
        /*neg_a=*/false, a, /*neg_b=*/false, b,
        /*c_mod=*/(short)0, c, /*reuse_a=*/false, /*reuse_b=*/false);
}

// ---------------------------------------------------------------------------
// Fill kernel (for -inf init etc.)
// ---------------------------------------------------------------------------
__global__ void fill_kernel(float* __restrict__ p, float val, long long n) {
    long long i = (long long)blockIdx.x * blockDim.x + threadIdx.x;
    if (i < n) p[i] = val;
}

// ---------------------------------------------------------------------------
// fp32 WMMA GEMM:  Y[M,N] = act( (A (+A2)) [M,K] @ W[K,N] + bias )
// Wave: 16 rows x 64 cols (4 accumulators). Block: 4 waves = 64 rows x 64 cols.
// A-layout per ISA: lane<16 -> {A[m,k0],A[m,k0+1]}, lane>=16 -> {A[m,k0+2],A[m,k0+3]}
// B-layout mirrored over N. C/D: vgpr r = D[r + 8*(lane/16), lane%16].
// Row clamp (not predication) keeps EXEC all-ones around WMMA.
// ---------------------------------------------------------------------------
template<bool HAS_A2, bool RELU>
__global__ void gemm_wmma_f32(const float* __restrict__ A,
                              const float* __restrict__ A2,
                              const float* __restrict__ Wt,
                              const float* __restrict__ bias,
                              float* __restrict__ Y,
                              int M, int K, int N)
{
    const int lane = threadIdx.x & 31;
    const int wave = threadIdx.x >> 5;
    const int rowBase = blockIdx.x * 64 + wave * 16;
    const int colBase = blockIdx.y * 64;
    const int half = lane >> 4;        // 0 or 1
    const int l16  = lane & 15;
    const int row  = rowBase + l16;
    const int rowc = row < M ? row : (M - 1);   // clamp so EXEC stays all-ones

    v8f acc0 = {0.f,0.f,0.f,0.f,0.f,0.f,0.f,0.f};
    v8f acc1 = acc0, acc2 = acc0, acc3 = acc0;

    // stride-bumped pointers: no per-iteration address multiplies
    const float* ap  = A + (size_t)rowc * K + 2 * half;
    const float* ap2 = HAS_A2 ? (A2 + (size_t)rowc * K + 2 * half) : nullptr;
    const float* wp  = Wt + (size_t)(2 * half) * N + colBase + l16;
    const int wstep = 4 * N;

    #pragma unroll 2
    for (int k0 = 0; k0 < K; k0 += 4, ap += 4, wp += wstep) {
        v2f a;
        a.x = ap[0];
        a.y = ap[1];
        if (HAS_A2) { a.x += ap2[0]; a.y += ap2[1]; ap2 += 4; }
        // 4 independent B fragments: loads clause together, drain under WMMAs
        v2f b0, b1, b2, b3;
        b0.x = wp[0];  b0.y = wp[N];
        b1.x = wp[16]; b1.y = wp[N + 16];
        b2.x = wp[32]; b2.y = wp[N + 32];
        b3.x = wp[48]; b3.y = wp[N + 48];
        acc0 = wmma_f32(a, b0, acc0);
        acc1 = wmma_f32(a, b1, acc1);
        acc2 = wmma_f32(a, b2, acc2);
        acc3 = wmma_f32(a, b3, acc3);
    }

    auto store_tile = [&](v8f acc, int tcol) {
        const int col = colBase + tcol + l16;
        const float bv = bias[col];
        #pragma unroll
        for (int r = 0; r < 8; ++r) {
            const int orow = rowBase + r + 8 * half;
            if (orow < M) {
                float v = acc[r] + bv;
                if (RELU) v = fmaxf(v, 0.0f);
                Y[(size_t)orow * N + col] = v;
            }
        }
    };
    store_tile(acc0, 0);
    store_tile(acc1, 16);
    store_tile(acc2, 32);
    store_tile(acc3, 48);
}

// ---------------------------------------------------------------------------
// GIN edge aggregation: agg[dst,:] += h[src,:]
// ---------------------------------------------------------------------------
__global__ void scatter_add_rows(const float* __restrict__ h,
                                 const int* __restrict__ src,
                                 const int* __restrict__ dst,
                                 float* __restrict__ agg,
                                 int nE, int F)
{
    long long idx = (long long)blockIdx.x * blockDim.x + threadIdx.x;
    long long total = (long long)nE * F;
    if (idx >= total) return;
    int e = (int)(idx / F);
    int f = (int)(idx % F);
    atomicAdd(&agg[(size_t)dst[e] * F + f], h[(size_t)src[e] * F + f]);
}

// ---------------------------------------------------------------------------
// float atomic max via int/uint trick (works with -inf init)
// ---------------------------------------------------------------------------
__device__ inline void atomicMaxFloat(float* addr, float val) {
    if (val >= 0.0f) {
        atomicMax((int*)addr, __float_as_int(val));
    } else {
        atomicMin((unsigned int*)addr, __float_as_uint(val));
    }
}

// ---------------------------------------------------------------------------
// GT pass 1: alpha_raw[e,h] = <q[dst,h,:], k[src,h,:] + e_proj[e,h,:]> / 8
// e_proj recomputed from edge_attr and LDS-cached We (16x128).
// Also atomic segment-max into m[dst,h].
// ---------------------------------------------------------------------------
__global__ void gt_edge_alpha(const float* __restrict__ q,
                              const float* __restrict__ k,
                              const float* __restrict__ ea,
                              const float* __restrict__ We,
                              const int* __restrict__ src,
                              const int* __restrict__ dst,
                              float* __restrict__ alpha,
                              float* __restrict__ mbuf,
                              int nE)
{
    __shared__ float sWe[16 * HC];
    for (int i = threadIdx.x; i < 16 * HC; i += blockDim.x) sWe[i] = We[i];
    __syncthreads();
    long long idx = (long long)blockIdx.x * blockDim.x + threadIdx.x;
    if (idx >= (long long)nE * 2) return;
    int e = (int)(idx >> 1);
    int h = (int)(idx & 1);
    int s = src[e], d = dst[e];
    float eav[16];
    #pragma unroll
    for (int j = 0; j < 16; ++j) eav[j] = ea[(size_t)e * 16 + j];
    const float* qr = q + (size_t)d * HC + h * HID;
    const float* kr = k + (size_t)s * HC + h * HID;
    const float* wcol = sWe + h * HID;
    float acc = 0.0f;
    for (int c = 0; c < HID; ++c) {
        float ep = 0.0f;
        #pragma unroll
        for (int j = 0; j < 16; ++j) ep += eav[j] * wcol[j * HC + c];
        acc += qr[c] * (kr[c] + ep);
    }
    acc *= 0.125f;   // 1/sqrt(64)
    alpha[idx] = acc;
    atomicMaxFloat(&mbuf[(size_t)d * 2 + h], acc);
}

// ---------------------------------------------------------------------------
// GT pass 2: ex = exp(alpha - m[dst]); s[dst,h]+=ex;
//            out[dst,h,:] += ex * (v[src,h,:] + e_proj)   (numerator)
// ---------------------------------------------------------------------------
__global__ void gt_edge_accum(const float* __restrict__ v,
                              const float* __restrict__ ea,
                              const float* __restrict__ We,
                              const int* __restrict__ src,
                              const int* __restrict__ dst,
                              const float* __restrict__ alpha,
                              const float* __restrict__ mbuf,
                              float* __restrict__ sbuf,
                              float* __restrict__ outb,
                              int nE)
{
    __shared__ float sWe[16 * HC];
    for (int i = threadIdx.x; i < 16 * HC; i += blockDim.x) sWe[i] = We[i];
    __syncthreads();
    long long idx = (long long)blockIdx.x * blockDim.x + threadIdx.x;
    if (idx >= (long long)nE * 2) return;
    int e = (int)(idx >> 1);
    int h = (int)(idx & 1);
    int s = src[e], d = dst[e];
    float ex = expf(alpha[idx] - mbuf[(size_t)d * 2 + h]);
    atomicAdd(&sbuf[(size_t)d * 2 + h], ex);
    float eav[16];
    #pragma unroll
    for (int j = 0; j < 16; ++j) eav[j] = ea[(size_t)e * 16 + j];
    const float* vr = v + (size_t)s * HC + h * HID;
    const float* wcol = sWe + h * HID;
    float* orow = outb + (size_t)d * HC + h * HID;
    for (int c = 0; c < HID; ++c) {
        float ep = 0.0f;
        #pragma unroll
        for (int j = 0; j < 16; ++j) ep += eav[j] * wcol[j * HC + c];
        atomicAdd(&orow[c], ex * (vr[c] + ep));
    }
}

// ---------------------------------------------------------------------------
// GT pass 3: normalize, gated skip:
//   o = out / (s + 1e-16); beta = sigmoid([o, xr, o-xr] @ betaW)
//   g = beta*xr + (1-beta)*o
// ---------------------------------------------------------------------------
__global__ void gt_node_finish(const float* __restrict__ outb,
                               const float* __restrict__ xr,
                               const float* __restrict__ sbuf,
                               const float* __restrict__ betaW,
                               float* __restrict__ gout,
                               int n)
{
    int i = blockIdx.x * blockDim.x + threadIdx.x;
    if (i >= n) return;
    float inv0 = 1.0f / (sbuf[(size_t)i * 2 + 0] + 1e-16f);
    float inv1 = 1.0f / (sbuf[(size_t)i * 2 + 1] + 1e-16f);
    const float* orow = outb + (size_t)i * HC;
    const float* xrow = xr + (size_t)i * HC;
    float bdot = 0.0f;
    for (int j = 0; j < HC; ++j) {
        float o = orow[j] * (j < HID ? inv0 : inv1);
        float x = xrow[j];
        bdot += o * betaW[j] + x * betaW[HC + j] + (o - x) * betaW[2 * HC + j];
    }
    float beta = 1.0f / (1.0f + expf(-bdot));
    float* grow = gout + (size_t)i * HC;
    for (int j = 0; j < HC; ++j) {
        float o = orow[j] * (j < HID ? inv0 : inv1);
        float x = xrow[j];
        grow[j] = beta * x + (1.0f - beta) * o;
    }
}

// ---------------------------------------------------------------------------
// Pooling: gin sum-pool (64 feats), gt sum-pool (128 feats) + node counts
// ---------------------------------------------------------------------------
__global__ void pool_kernel(const float* __restrict__ h,
                            const float* __restrict__ g,
                            const int* __restrict__ batch,
                            float* __restrict__ ginpool,
                            float* __restrict__ gtpool,
                            float* __restrict__ cnt,
                            int n)
{
    long long idx = (long long)blockIdx.x * blockDim.x + threadIdx.x;
    if (idx >= (long long)n * HC) return;
    int i = (int)(idx / HC);
    int f = (int)(idx % HC);
    int b = batch[i];
    if (f < HID) atomicAdd(&ginpool[(size_t)b * HID + f], h[(size_t)i * HID + f]);
    atomicAdd(&gtpool[(size_t)b * HC + f], g[(size_t)i * HC + f]);
    if (f == 0) atomicAdd(&cnt[b], 1.0f);
}

// ---------------------------------------------------------------------------
// Readout: emb=[gin_pool, gt_pool/cnt] (192) -> relu(192x64) -> 64x1
// ---------------------------------------------------------------------------
__global__ void readout_kernel(const float* __restrict__ ginpool,
                               const float* __restrict__ gtpool,
                               const float* __restrict__ cnt,
                               const float* __restrict__ w1,
                               const float* __restrict__ b1,
                               const float* __restrict__ w2,
                               const float* __restrict__ b2,
                               float* __restrict__ out)
{
    int g = threadIdx.x;
    if (g >= NGRAPH) return;
    float invc = 1.0f / fmaxf(cnt[g], 1.0f);
    float hid[HID];
    for (int o = 0; o < HID; ++o) {
        float acc = b1[o];
        for (int j = 0; j < HID; ++j)
            acc += ginpool[(size_t)g * HID + j] * w1[(size_t)j * HID + o];
        for (int j = 0; j < HC; ++j)
            acc += gtpool[(size_t)g * HC + j] * invc * w1[(size_t)(HID + j) * HID + o];
        hid[o] = fmaxf(acc, 0.0f);
    }
    float res = b2[0];
    for (int o = 0; o < HID; ++o) res += hid[o] * w2[o];
    out[g] = res;
}

// ---------------------------------------------------------------------------
// Host orchestration
// ---------------------------------------------------------------------------
static inline void launch_gemm(const float* A, const float* A2, const float* Wt,
                               const float* bias, float* Y, int M, int K, int N,
                               int relu, hipStream_t stream)
{
    dim3 grid((M + 63) / 64, N / 64);
    dim3 block(128);
    if (A2) {
        if (relu) hipLaunchKernelGGL((gemm_wmma_f32<true,  true >), grid, block, 0, stream, A, A2, Wt, bias, Y, M, K, N);
        else      hipLaunchKernelGGL((gemm_wmma_f32<true,  false>), grid, block, 0, stream, A, A2, Wt, bias, Y, M, K, N);
    } else {
        if (relu) hipLaunchKernelGGL((gemm_wmma_f32<false, true >), grid, block, 0, stream, A, A2, Wt, bias, Y, M, K, N);
        else      hipLaunchKernelGGL((gemm_wmma_f32<false, false>), grid, block, 0, stream, A, A2, Wt, bias, Y, M, K, N);
    }
}

extern "C" void kernel_launch(void* const* d_in, const int* in_sizes, int n_in,
                              void* d_out, int out_size, void* d_ws, size_t ws_size,
                              hipStream_t stream)
{
    const float* x    = (const float*)d_in[0];
    const float* eatt = (const float*)d_in[1];
    const int*   ei   = (const int*)d_in[2];
    const int*   batch= (const int*)d_in[3];

    const int n  = in_sizes[0] / NFEAT;     // 25000
    const int E  = in_sizes[2] / 2;         // 400000
    const int* src = ei;
    const int* dst = ei + E;

    // param leaves
    const float* gin_l1w[3] = {(const float*)d_in[4],  (const float*)d_in[8],  (const float*)d_in[12]};
    const float* gin_l1b[3] = {(const float*)d_in[5],  (const float*)d_in[9],  (const float*)d_in[13]};
    const float* gin_l2w[3] = {(const float*)d_in[6],  (const float*)d_in[10], (const float*)d_in[14]};
    const float* gin_l2b[3] = {(const float*)d_in[7],  (const float*)d_in[11], (const float*)d_in[15]};
    const float* gt_qw[2]   = {(const float*)d_in[16], (const float*)d_in[26]};
    const float* gt_qb[2]   = {(const float*)d_in[17], (const float*)d_in[27]};
    const float* gt_kw[2]   = {(const float*)d_in[18], (const float*)d_in[28]};
    const float* gt_kb[2]   = {(const float*)d_in[19], (const float*)d_in[29]};
    const float* gt_vw[2]   = {(const float*)d_in[20], (const float*)d_in[30]};
    const float* gt_vb[2]   = {(const float*)d_in[21], (const float*)d_in[31]};
    const float* gt_ew[2]   = {(const float*)d_in[22], (const float*)d_in[32]};
    const float* gt_sw[2]   = {(const float*)d_in[23], (const float*)d_in[33]};
    const float* gt_sb[2]   = {(const float*)d_in[24], (const float*)d_in[34]};
    const float* gt_bw[2]   = {(const float*)d_in[25], (const float*)d_in[35]};
    const float* r1w = (const float*)d_in[36];
    const float* r1b = (const float*)d_in[37];
    const float* r2w = (const float*)d_in[38];
    const float* r2b = (const float*)d_in[39];

    // workspace carve-up (floats)
    float* W = (float*)d_ws;
    size_t o = 0;
    auto alloc = [&](size_t cnt_) { float* p = W + o; o += cnt_; return p; };
    float* agg   = alloc((size_t)n * NFEAT);
    float* t1    = alloc((size_t)n * HID);
    float* hA    = alloc((size_t)n * HID);
    float* hB    = alloc((size_t)n * HID);
    float* gA    = alloc((size_t)n * HC);
    float* gB    = alloc((size_t)n * HC);
    float* qb    = alloc((size_t)n * HC);
    float* kb    = alloc((size_t)n * HC);
    float* vb    = alloc((size_t)n * HC);
    float* xrb   = alloc((size_t)n * HC);
    float* outb  = alloc((size_t)n * HC);
    float* alphab= alloc((size_t)E * 2);
    float* mb    = alloc((size_t)n * 2);
    float* sb    = alloc((size_t)n * 2);
    float* ginpool = alloc((size_t)NGRAPH * HID);
    float* gtpool  = alloc((size_t)NGRAPH * HC);
    float* cntb    = alloc((size_t)NGRAPH);
    (void)ws_size; (void)n_in; (void)out_size;

    const int TB = 256;
    auto blocks = [](long long tot, int tb) { return (unsigned)((tot + tb - 1) / tb); };

    // ================= GIN branch =================
    // layer 0: F=128 in, 64 out
    hipMemsetAsync(agg, 0, (size_t)n * NFEAT * sizeof(float), stream);
    hipLaunchKernelGGL(scatter_add_rows, dim3(blocks((long long)E * NFEAT, TB)), dim3(TB), 0, stream,
                       x, src, dst, agg, E, NFEAT);
    launch_gemm(x, agg, gin_l1w[0], gin_l1b[0], t1, n, NFEAT, HID, 1, stream);
    launch_gemm(t1, nullptr, gin_l2w[0], gin_l2b[0], hA, n, HID, HID, 0, stream);
    // layer 1
    hipMemsetAsync(agg, 0, (size_t)n * HID * sizeof(float), stream);
    hipLaunchKernelGGL(scatter_add_rows, dim3(blocks((long long)E * HID, TB)), dim3(TB), 0, stream,
                       hA, src, dst, agg, E, HID);
    launch_gemm(hA, agg, gin_l1w[1], gin_l1b[1], t1, n, HID, HID, 1, stream);
    launch_gemm(t1, nullptr, gin_l2w[1], gin_l2b[1], hB, n, HID, HID, 0, stream);
    // layer 2
    hipMemsetAsync(agg, 0, (size_t)n * HID * sizeof(float), stream);
    hipLaunchKernelGGL(scatter_add_rows, dim3(blocks((long long)E * HID, TB)), dim3(TB), 0, stream,
                       hB, src, dst, agg, E, HID);
    launch_gemm(hB, agg, gin_l1w[2], gin_l1b[2], t1, n, HID, HID, 1, stream);
    launch_gemm(t1, nullptr, gin_l2w[2], gin_l2b[2], hA, n, HID, HID, 0, stream);
    // final GIN features: hA

    // ================= TransformerConv branch =================
    const float* gin_feat[2] = {x, gA};
    float* gout_feat[2] = {gA, gB};
    for (int L = 0; L < 2; ++L) {
        const float* g = gin_feat[L];
        float* gnext = gout_feat[L];
        launch_gemm(g, nullptr, gt_qw[L], gt_qb[L], qb,  n, NFEAT, HC, 0, stream);
        launch_gemm(g, nullptr, gt_kw[L], gt_kb[L], kb,  n, NFEAT, HC, 0, stream);
        launch_gemm(g, nullptr, gt_vw[L], gt_vb[L], vb,  n, NFEAT, HC, 0, stream);
        launch_gemm(g, nullptr, gt_sw[L], gt_sb[L], xrb, n, NFEAT, HC, 0, stream);

        hipLaunchKernelGGL(fill_kernel, dim3(blocks((long long)n * 2, TB)), dim3(TB), 0, stream,
                           mb, -INFINITY, (long long)n * 2);
        hipMemsetAsync(sb, 0, (size_t)n * 2 * sizeof(float), stream);
        hipMemsetAsync(outb, 0, (size_t)n * HC * sizeof(float), stream);

        hipLaunchKernelGGL(gt_edge_alpha, dim3(blocks((long long)E * 2, TB)), dim3(TB), 0, stream,
                           qb, kb, eatt, gt_ew[L], src, dst, alphab, mb, E);
        hipLaunchKernelGGL(gt_edge_accum, dim3(blocks((long long)E * 2, TB)), dim3(TB), 0, stream,
                           vb, eatt, gt_ew[L], src, dst, alphab, mb, sb, outb, E);
        hipLaunchKernelGGL(gt_node_finish, dim3(blocks(n, 128)), dim3(128), 0, stream,
                           outb, xrb, sb, gt_bw[L], gnext, n);
    }
    // final GT features: gB

    // ================= Pool + readout =================
    hipMemsetAsync(ginpool, 0, (size_t)NGRAPH * HID * sizeof(float), stream);
    hipMemsetAsync(gtpool,  0, (size_t)NGRAPH * HC  * sizeof(float), stream);
    hipMemsetAsync(cntb,    0, (size_t)NGRAPH * sizeof(float), stream);
    hipLaunchKernelGGL(pool_kernel, dim3(blocks((long long)n * HC, TB)), dim3(TB), 0, stream,
                       hA, gB, batch, ginpool, gtpool, cntb, n);
    hipLaunchKernelGGL(readout_kernel, dim3(1), dim3(64), 0, stream,
                       ginpool, gtpool, cntb, r1w, r1b, r2w, r2b, (float*)d_out);
}